// Graph_MPS_RNN_11596411699491
// MI455X (gfx1250) — compile-verified
//
#include <hip/hip_runtime.h>
#include <math.h>

// Problem constants (match reference)
#define NB   16384
#define NQ   100
#define NS   50
#define DCUT 32
#define HLOC 4

#define MPAD 34   // padded row stride (floats) for M in LDS -> conflict-free A loads
#define HPAD 17   // padded column stride for per-wave h shuffle buffer

typedef __attribute__((ext_vector_type(2))) float v2f;
typedef __attribute__((ext_vector_type(8))) float v8f;

static __device__ __forceinline__ v8f wmma4(v2f a, v2f b, v8f c) {
    // D = A(16x4 f32) * B(4x16 f32) + C(16x16 f32)
    return __builtin_amdgcn_wmma_f32_16x16x4_f32(
        /*neg_a=*/false, a, /*neg_b=*/false, b,
        /*c_mod=*/(short)0, c, /*reuse_a=*/false, /*reuse_b=*/false);
}

__global__ __launch_bounds__(128) void mps_rnn_wmma_kernel(
    const int*   __restrict__ xg,   // [NB, NQ]
    const float* __restrict__ Mg,   // [NS, HLOC, DCUT, DCUT, 2]
    const float* __restrict__ vg,   // [NS, HLOC, DCUT, 2]
    const float* __restrict__ wg,   // [NS, DCUT, 2]
    const float* __restrict__ cg,   // [NS, 2]
    float*       __restrict__ outg) // [NB, 2]
{
    __shared__ float Mr[HLOC * DCUT * MPAD];
    __shared__ float Mi[HLOC * DCUT * MPAD];
    __shared__ float vr_s[HLOC * DCUT];
    __shared__ float vi_s[HLOC * DCUT];
    __shared__ float wr_s[DCUT];
    __shared__ float wi_s[DCUT];
    __shared__ float c_s[2];
    __shared__ float hbuf[4][2][DCUT][HPAD];  // per-wave h shuffle buffer

    const int tid  = threadIdx.x;
    const int lane = tid & 31;
    const int wave = tid >> 5;
    const int n    = lane & 15;   // batch column within the wave's 16-wide tile
    const int half = lane >> 4;   // which half-wave (selects K/M sub-rows)
    const int bcol = blockIdx.x * 64 + wave * 16 + n;

    // Hidden state as WMMA B operands: Hr/Hi[kk] covers K rows kk*4..kk*4+3.
    // Per-lane B layout: lanes hold N=n; vgpr0 = K row (kk*4 + 2*half), vgpr1 = next row.
    v2f Hr[8], Hi[8];
    const float h0 = 0.17677669529663688f;  // 1/sqrt(32)
    #pragma unroll
    for (int k = 0; k < 8; ++k) {
        Hr[k][0] = h0; Hr[k][1] = h0;
        Hi[k][0] = 0.f; Hi[k][1] = 0.f;
    }

    float logp = 0.f, phase = 0.f;
    const int k0h = half * 2;  // A-operand K sub-offset per half-wave

    for (int t = 0; t < NS; ++t) {
        // ---------- stage M / v / w / c for this site into LDS ----------
        {
            const float* Mt = Mg + (size_t)t * HLOC * DCUT * DCUT * 2;
            for (int idx = tid; idx < HLOC * DCUT * DCUT; idx += 128) {
                const int s = idx >> 10, rem = idx & 1023;
                const int d = rem >> 5,  e  = rem & 31;
                const float re = Mt[idx * 2 + 0];
                const float im = Mt[idx * 2 + 1];
                Mr[(s * DCUT + d) * MPAD + e] = re;
                Mi[(s * DCUT + d) * MPAD + e] = im;
            }
            const float* vt = vg + (size_t)t * HLOC * DCUT * 2;
            if (tid < HLOC * DCUT) { vr_s[tid] = vt[tid * 2]; vi_s[tid] = vt[tid * 2 + 1]; }
            const float* wt = wg + (size_t)t * DCUT * 2;
            if (tid < DCUT) { wr_s[tid] = wt[tid * 2]; wi_s[tid] = wt[tid * 2 + 1]; }
            if (tid == 0) { c_s[0] = cg[t * 2]; c_s[1] = cg[t * 2 + 1]; }
            if (t + 1 < NS) {  // pull next site's 32KB M tile toward L2/L0
                const char* nxt = (const char*)(Mg + (size_t)(t + 1) * HLOC * DCUT * DCUT * 2);
                __builtin_prefetch(nxt + tid * 256, 0, 1);
            }
        }
        __syncthreads();

        // ---------- ht[sigma] = M[sigma] @ H + v[sigma]  (complex, WMMA) ----------
        v8f htR[HLOC][2], htI[HLOC][2];
        float p_part[HLOC];
        #pragma unroll
        for (int s = 0; s < HLOC; ++s) {
            p_part[s] = 0.f;
            #pragma unroll
            for (int mt = 0; mt < 2; ++mt) {
                v8f accR  = {};
                v8f accMI = {};
                v8f accI  = {};
                const float* mrBase = &Mr[((s * DCUT) + mt * 16 + n) * MPAD];
                const float* miBase = &Mi[((s * DCUT) + mt * 16 + n) * MPAD];
                #pragma unroll
                for (int kk = 0; kk < 8; ++kk) {
                    const int koff = kk * 4 + k0h;  // even -> 8B aligned
                    const v2f Ar = *(const v2f*)(mrBase + koff);
                    const v2f Ai = *(const v2f*)(miBase + koff);
                    accR  = wmma4(Ar, Hr[kk], accR);   // Mr*Hr
                    accMI = wmma4(Ai, Hi[kk], accMI);  // Mi*Hi (subtract later)
                    accI  = wmma4(Ar, Hi[kk], accI);   // Mr*Hi
                    accI  = wmma4(Ai, Hr[kk], accI);   // + Mi*Hr
                }
                #pragma unroll
                for (int r = 0; r < 8; ++r) {
                    const int row = mt * 16 + r + half * 8;  // D-layout row of this vgpr
                    const float hr = accR[r] - accMI[r] + vr_s[s * DCUT + row];
                    const float hi = accI[r]            + vi_s[s * DCUT + row];
                    htR[s][mt][r] = hr;
                    htI[s][mt][r] = hi;
                    p_part[s] += hr * hr + hi * hi;
                }
            }
        }

        // ---------- Born probabilities per batch column ----------
        float p_sig[HLOC], psum = 0.f;
        #pragma unroll
        for (int s = 0; s < HLOC; ++s) {
            p_sig[s] = p_part[s] + __shfl_xor(p_part[s], 16, 32);
            psum += p_sig[s];
        }

        // local state index for this batch column at this site
        const int xs0 = xg[(size_t)bcol * NQ + 2 * t];
        const int xs1 = xg[(size_t)bcol * NQ + 2 * t + 1];
        const int ss  = 2 * xs0 + xs1;

        const float p_sel = (ss == 0) ? p_sig[0] : (ss == 1) ? p_sig[1]
                          : (ss == 2) ? p_sig[2] : p_sig[3];
        logp += logf(p_sel) - logf(psum);
        const float rinv = rsqrtf(p_sel);

        // ---------- select sigma, renormalize, phase amplitude, stash h ----------
        float ampR = 0.f, ampI = 0.f;
        #pragma unroll
        for (int mt = 0; mt < 2; ++mt) {
            const v8f selR = (ss == 0) ? htR[0][mt] : (ss == 1) ? htR[1][mt]
                           : (ss == 2) ? htR[2][mt] : htR[3][mt];
            const v8f selI = (ss == 0) ? htI[0][mt] : (ss == 1) ? htI[1][mt]
                           : (ss == 2) ? htI[2][mt] : htI[3][mt];
            #pragma unroll
            for (int r = 0; r < 8; ++r) {
                const int row = mt * 16 + r + half * 8;
                const float hr = selR[r] * rinv;
                const float hi = selI[r] * rinv;
                hbuf[wave][0][row][n] = hr;
                hbuf[wave][1][row][n] = hi;
                const float wr = wr_s[row], wi = wi_s[row];
                ampR += wr * hr - wi * hi;
                ampI += wr * hi + wi * hr;
            }
        }
        ampR += __shfl_xor(ampR, 16, 32);
        ampI += __shfl_xor(ampI, 16, 32);
        ampR += c_s[0];
        ampI += c_s[1];
        phase += atan2f(ampI, ampR);

        __syncthreads();  // all M reads done; safe to restage next site after reload

        // ---------- reload h as B operands for next site ----------
        #pragma unroll
        for (int kk = 0; kk < 8; ++kk) {
            const int krow = kk * 4 + half * 2;
            Hr[kk][0] = hbuf[wave][0][krow + 0][n];
            Hr[kk][1] = hbuf[wave][0][krow + 1][n];
            Hi[kk][0] = hbuf[wave][1][krow + 0][n];
            Hi[kk][1] = hbuf[wave][1][krow + 1][n];
        }
    }

    // ---------- write [0.5*logp, phase] ----------
    if (half == 0) {
        float2 o;
        o.x = 0.5f * logp;
        o.y = phase;
        *(float2*)(outg + (size_t)bcol * 2) = o;
    }
}

extern "C" void kernel_launch(void* const* d_in, const int* in_sizes, int n_in,
                              void* d_out, int out_size, void* d_ws, size_t ws_size,
                              hipStream_t stream) {
    (void)in_sizes; (void)n_in; (void)out_size; (void)d_ws; (void)ws_size;
    const int*   x  = (const int*)  d_in[0];
    const float* M  = (const float*)d_in[1];
    const float* v  = (const float*)d_in[2];
    const float* w  = (const float*)d_in[3];
    const float* c  = (const float*)d_in[4];
    float* out = (float*)d_out;

    dim3 grid(NB / 64);   // 256 blocks, each block = 4 waves x 16 batch columns
    dim3 block(128);
    mps_rnn_wmma_kernel<<<grid, block, 0, stream>>>(x, M, v, w, c, out);
}